// ParametricMemoryModel_3204045603471
// MI455X (gfx1250) — compile-verified
//
#include <hip/hip_runtime.h>
#include <hip/hip_bf16.h>
#include <math.h>

typedef __attribute__((ext_vector_type(2))) float v2f;
typedef __attribute__((ext_vector_type(8))) float v8f;

#define LRC  0.01f
#define EPSC 1e-5f

#define Bq 256
#define Lq 4096
#define Hq 32
#define Vq 64
#define CH 16
#define NCH (Lq / CH)      // 256 chunks of 16 tokens
#define A1STR 68           // padded stride for 16x64 relu buffer (bank-conflict-free)
#define HSTR  36           // padded stride for 16x32 hidden buffer

__launch_bounds__(64, 1)
__global__ void pmm_fused_kernel(const int* __restrict__ seq,
                                 const float* __restrict__ embed,
                                 const float* __restrict__ ff_w1,
                                 const float* __restrict__ ff_b1,
                                 const float* __restrict__ ff_w2,
                                 const float* __restrict__ ff_b2,
                                 const float* __restrict__ ln_g,
                                 const float* __restrict__ ln_b,
                                 const float* __restrict__ mlp_w1,
                                 const float* __restrict__ mlp_b1,
                                 const float* __restrict__ mlp_w2,
                                 const float* __restrict__ mlp_b2,
                                 const float* __restrict__ out_w,
                                 const float* __restrict__ out_b,
                                 float* __restrict__ out)
{
    __shared__ float embedLds[Vq * Hq];      // 64 x 32
    __shared__ float w1Lds[2 * Hq * Hq];     // ff_w1: 64 x 32
    __shared__ float w2Lds[Hq * 2 * Hq];     // ff_w2: 32 x 64
    __shared__ float a1buf[CH * A1STR];      // relu(fc1) staging, 16 x 64 (padded)
    __shared__ float hid[2][CH * HSTR];      // hidden double buffer, 16 x 32 (padded)
    __shared__ float ctxLds[Hq];

    const int b    = blockIdx.x;
    const int tid  = threadIdx.x;
    const int lane = tid & 31;
    const int wave = tid >> 5;
    const int m    = lane & 15;   // M/N index within a 16-wide tile
    const int hi   = lane >> 4;   // K half selector for f32 WMMA operands

    // ---- cooperative load of shared tables (all L2-resident, tiny) ----
    for (int i = tid; i < Vq * Hq; i += 64)     embedLds[i] = embed[i];
    for (int i = tid; i < 2 * Hq * Hq; i += 64) w1Lds[i]    = ff_w1[i];
    for (int i = tid; i < 2 * Hq * Hq; i += 64) w2Lds[i]    = ff_w2[i];
    __syncthreads();

    // ---- producer-wave constants ----
    float b1t[4], b2t[2], lnGv = 0.f, lnBv = 0.f;
    if (wave == 0) {
        #pragma unroll
        for (int nt = 0; nt < 4; ++nt) b1t[nt] = ff_b1[nt * 16 + m];
        #pragma unroll
        for (int nt = 0; nt < 2; ++nt) b2t[nt] = ff_b2[nt * 16 + m];
        lnGv = ln_g[lane];
        lnBv = ln_b[lane];
    }

    // ---- consumer-wave per-sample fast-weight state (lane = h) ----
    float w1c[8], b1v[8], w2r[8], b2v = 0.f;
    if (wave == 1) {
        #pragma unroll
        for (int i = 0; i < 8; ++i) {
            w1c[i] = mlp_w1[i * Hq + lane];   // w1[i, h]
            b1v[i] = mlp_b1[i];               // replicated
            w2r[i] = mlp_w2[lane * 8 + i];    // w2[h, i]
        }
        b2v = mlp_b2[lane];
    }

    // ================= producer: encode one 16-token chunk =================
    auto produce = [&](int c, float* hbuf) {
        const int base = c * CH;
        const int tok  = seq[b * Lq + base + m];   // lane holds token id for row m

        // A operands for FF1: E[16,32], lane m, K pair (4kb + 2hi, +1)
        v2f eA[8];
        #pragma unroll
        for (int kb = 0; kb < 8; ++kb)
            eA[kb] = *(const v2f*)&embedLds[tok * Hq + 4 * kb + 2 * hi];

        // FF1: Z1[16,64] = E @ ff_w1^T  (B[k][n] = ff_w1[n][k])
        #pragma unroll 1
        for (int nt = 0; nt < 4; ++nt) {
            const int n0 = nt * 16;
            v8f acc = {};
            #pragma unroll
            for (int kb = 0; kb < 8; ++kb) {
                v2f bB = *(const v2f*)&w1Lds[(n0 + m) * Hq + 4 * kb + 2 * hi];
                acc = __builtin_amdgcn_wmma_f32_16x16x4_f32(
                        false, eA[kb], false, bB, (short)0, acc, false, false);
            }
            const float bias = b1t[nt];
            #pragma unroll
            for (int r = 0; r < 8; ++r) {           // D: row r + 8*hi, col n0+m
                float z = acc[r] + bias;
                a1buf[(r + 8 * hi) * A1STR + n0 + m] = fmaxf(z, 0.f);
            }
        }
        asm volatile("s_wait_dscnt 0" ::: "memory");   // intra-wave LDS transpose RAW
        __builtin_amdgcn_wave_barrier();

        // A operands for FF2: relu output [16,64]
        v2f a2[16];
        #pragma unroll
        for (int kb = 0; kb < 16; ++kb)
            a2[kb] = *(const v2f*)&a1buf[m * A1STR + 4 * kb + 2 * hi];

        // FF2: Z2[16,32] = A1 @ ff_w2^T  (B[k][n] = ff_w2[n][k]); + bias + residual
        #pragma unroll 1
        for (int nt = 0; nt < 2; ++nt) {
            const int n0 = nt * 16;
            v8f acc = {};
            #pragma unroll
            for (int kb = 0; kb < 16; ++kb) {
                v2f bB = *(const v2f*)&w2Lds[(n0 + m) * (2 * Hq) + 4 * kb + 2 * hi];
                acc = __builtin_amdgcn_wmma_f32_16x16x4_f32(
                        false, a2[kb], false, bB, (short)0, acc, false, false);
            }
            const float bias = b2t[nt];
            #pragma unroll
            for (int r = 0; r < 8; ++r) {
                int   tm = __shfl(tok, r + 8 * hi, 32);            // token id of row r+8hi
                float e  = embedLds[tm * Hq + n0 + m];             // residual
                hbuf[(r + 8 * hi) * HSTR + n0 + m] = acc[r] + bias + e;
            }
        }
        asm volatile("s_wait_dscnt 0" ::: "memory");
        __builtin_amdgcn_wave_barrier();

        // LayerNorm over H=32 per token (lane = h), in-place
        #pragma unroll 1
        for (int t = 0; t < CH; ++t) {
            float z = hbuf[t * HSTR + lane];
            float s = z, s2 = z * z;
            #pragma unroll
            for (int off = 16; off >= 1; off >>= 1) {
                s  += __shfl_xor(s,  off, 32);
                s2 += __shfl_xor(s2, off, 32);
            }
            float mu  = s * (1.f / 32.f);
            float var = s2 * (1.f / 32.f) - mu * mu;
            float nz  = (z - mu) * rsqrtf(var + EPSC);
            hbuf[t * HSTR + lane] = nz * lnGv + lnBv;
        }
        asm volatile("s_wait_dscnt 0" ::: "memory");
    };

    // ================= consumer: sequential SGD steps (lane = h) =================
    auto consume = [&](const float* hbuf, int nsteps) {
        #pragma unroll 1
        for (int s = 0; s < nsteps; ++s) {
            float x = hbuf[(2 * s) * HSTR + lane];
            float v = hbuf[(2 * s + 1) * HSTR + lane];

            float p[8];
            #pragma unroll
            for (int i = 0; i < 8; ++i) p[i] = w1c[i] * x;
            #pragma unroll
            for (int off = 16; off >= 1; off >>= 1) {
                #pragma unroll
                for (int i = 0; i < 8; ++i) p[i] += __shfl_xor(p[i], off, 32);
            }
            float a1p[8], a1v[8];
            float y = b2v;
            #pragma unroll
            for (int i = 0; i < 8; ++i) {
                a1p[i] = p[i] + b1v[i];
                a1v[i] = fmaxf(a1p[i], 0.f);
                y = fmaf(w2r[i], a1v[i], y);
            }
            float d = (y - v) * 0.0625f;            // dL/dy, mean over 32 outputs

            float q[8];                              // grad wrt a1 (pre-update w2)
            #pragma unroll
            for (int i = 0; i < 8; ++i) q[i] = w2r[i] * d;
            #pragma unroll
            for (int off = 16; off >= 1; off >>= 1) {
                #pragma unroll
                for (int i = 0; i < 8; ++i) q[i] += __shfl_xor(q[i], off, 32);
            }
            b2v -= LRC * d;
            #pragma unroll
            for (int i = 0; i < 8; ++i) {
                float g1 = (a1p[i] > 0.f) ? q[i] : 0.f;
                w2r[i] = fmaf(-LRC * d, a1v[i], w2r[i]);
                b1v[i] -= LRC * g1;
                w1c[i] = fmaf(-LRC * g1, x, w1c[i]);
            }
        }
    };

    // ================= pipelined main loop =================
    if (wave == 0) produce(0, hid[0]);
    __syncthreads();

    for (int c = 0; c < NCH; ++c) {
        if (wave == 0) {
            if (c + 1 < NCH) produce(c + 1, hid[(c + 1) & 1]);
        } else {
            consume(hid[c & 1], (c == NCH - 1) ? 7 : 8);  // ts = 0..4092 step 2
        }
        __syncthreads();
    }

    // final inference on hidden[L-1] (row 15 of last chunk, buffer 1)
    if (wave == 1) {
        const float* hbuf = hid[(NCH - 1) & 1];
        float x = hbuf[15 * HSTR + lane];
        float p[8];
        #pragma unroll
        for (int i = 0; i < 8; ++i) p[i] = w1c[i] * x;
        #pragma unroll
        for (int off = 16; off >= 1; off >>= 1) {
            #pragma unroll
            for (int i = 0; i < 8; ++i) p[i] += __shfl_xor(p[i], off, 32);
        }
        float y = b2v;
        #pragma unroll
        for (int i = 0; i < 8; ++i) y = fmaf(w2r[i], fmaxf(p[i] + b1v[i], 0.f), y);
        ctxLds[lane] = y;
    }
    __syncthreads();

    // out[b, v] = ctx . out_w[v, :] + out_b[v], one output per thread (v = tid)
    {
        float acc = out_b[tid];
        #pragma unroll
        for (int h = 0; h < Hq; ++h)
            acc = fmaf(ctxLds[h], out_w[tid * Hq + h], acc);
        out[b * Vq + tid] = acc;
    }
}

extern "C" void kernel_launch(void* const* d_in, const int* in_sizes, int n_in,
                              void* d_out, int out_size, void* d_ws, size_t ws_size,
                              hipStream_t stream) {
    (void)in_sizes; (void)n_in; (void)out_size; (void)d_ws; (void)ws_size;
    const int*   seq    = (const int*)  d_in[0];
    const float* embed  = (const float*)d_in[1];
    const float* ff_w1  = (const float*)d_in[2];
    const float* ff_b1  = (const float*)d_in[3];
    const float* ff_w2  = (const float*)d_in[4];
    const float* ff_b2  = (const float*)d_in[5];
    const float* ln_g   = (const float*)d_in[6];
    const float* ln_b   = (const float*)d_in[7];
    const float* mlp_w1 = (const float*)d_in[8];
    const float* mlp_b1 = (const float*)d_in[9];
    const float* mlp_w2 = (const float*)d_in[10];
    const float* mlp_b2 = (const float*)d_in[11];
    const float* out_w  = (const float*)d_in[12];
    const float* out_b  = (const float*)d_in[13];
    float*       out    = (float*)d_out;

    pmm_fused_kernel<<<dim3(Bq), dim3(64), 0, stream>>>(
        seq, embed, ff_w1, ff_b1, ff_w2, ff_b2, ln_g, ln_b,
        mlp_w1, mlp_b1, mlp_w2, mlp_b2, out_w, out_b, out);
}